// GaussianWindow_51135880626727
// MI455X (gfx1250) — compile-verified
//
#include <hip/hip_runtime.h>

namespace {
constexpr int B_ = 8, T_ = 1024, D_ = 400, K_ = 10, U_ = 1024, V_ = 80;
}

typedef __attribute__((ext_vector_type(2))) float v2f;
typedef __attribute__((ext_vector_type(8))) float v8f;

// Stash layout inside each phi output row (overwritten later by real phi):
//   prow[0..9]   alpha = exp(x@Wa+ba)
//   prow[10..19] beta  = exp(x@Wb+bb)
//   prow[20..29] exp(x@Wk+bk), then in-place inclusive cumsum over t -> kappa

// ---------------------------------------------------------------- projections
__global__ void __launch_bounds__(32)
proj_kernel(const float* __restrict__ x,
            const float* __restrict__ Wa, const float* __restrict__ ba,
            const float* __restrict__ Wb, const float* __restrict__ bb,
            const float* __restrict__ Wk, const float* __restrict__ bk,
            float* __restrict__ phi)
{
    __shared__ float xs[D_];
    const int bt = blockIdx.x;                       // b*T + t
    const float* xrow = x + (size_t)bt * D_;
    for (int i = threadIdx.x; i < D_; i += 32) xs[i] = xrow[i];
    __syncthreads();

    const int lane = threadIdx.x;
    if (lane < 3 * K_) {
        const int p = lane / K_;
        const int k = lane - p * K_;
        const float* W    = (p == 0) ? Wa : (p == 1) ? Wb : Wk;
        const float* bias = (p == 0) ? ba : (p == 1) ? bb : bk;
        float acc = bias[k];
        #pragma unroll 4
        for (int d = 0; d < D_; ++d) acc += xs[d] * W[d * K_ + k];
        phi[(size_t)bt * U_ + lane] = __expf(acc);
    }
}

// ------------------------------------------------------- cumsum over T (kappa)
__global__ void __launch_bounds__(1024)
scan_kernel(float* __restrict__ phi)
{
    __shared__ float buf[T_];
    const int b = blockIdx.x / K_;
    const int k = blockIdx.x - b * K_;
    const int t = threadIdx.x;
    const size_t idx = ((size_t)(b * T_ + t)) * U_ + 2 * K_ + k;

    float v = phi[idx];
    buf[t] = v;
    __syncthreads();
    // Hillis-Steele inclusive scan over 1024 elements
    for (int off = 1; off < T_; off <<= 1) {
        const float other = (t >= off) ? buf[t - off] : 0.0f;
        __syncthreads();
        v += other;
        buf[t] = v;
        __syncthreads();
    }
    phi[idx] = v;
}

// --------------------------------------------------------------------- phi
__global__ void __launch_bounds__(256)
phi_kernel(float* __restrict__ phi)
{
    __shared__ float pa[K_], pb[K_], pk[K_];
    const int bt  = blockIdx.x;                      // b*T + t
    float* prow   = phi + (size_t)bt * U_;
    const int tid = threadIdx.x;

    if (tid < 3 * K_) {
        const float v = prow[tid];
        if (tid < K_)          pa[tid]          = v;
        else if (tid < 2 * K_) pb[tid - K_]     = v;
        else                   pk[tid - 2 * K_] = v;
    }
    __syncthreads();   // stash now cached in LDS; safe to overwrite prow

    #pragma unroll
    for (int j = 0; j < U_ / 256; ++j) {
        const int   u  = tid + 256 * j;
        const float fu = (float)u;
        float s = 0.0f;
        #pragma unroll
        for (int k = 0; k < K_; ++k) {
            const float d = pk[k] - fu;
            s += pa[k] * __expf(-pb[k] * d * d);     // arg <= 0: no overflow
        }
        prow[u] = s;
    }
}

// -------------------------------------------- window = phi @ onehot via WMMA
// One wave computes a 16(M) x 80(N) tile of window[b], K = U = 1024 in steps
// of 4 using V_WMMA_F32_16X16X4_F32. A fragment reused across all 5 N-tiles.
__global__ void __launch_bounds__(128)
window_kernel(const float* __restrict__ phi, const float* __restrict__ onehot,
              float* __restrict__ win)
{
    const int wave = threadIdx.x >> 5;
    const int lane = threadIdx.x & 31;
    const int half = lane >> 4;          // 0: lanes 0-15, 1: lanes 16-31
    const int lid  = lane & 15;
    const int tile = blockIdx.x * 4 + wave;          // 0 .. B*T/16 - 1
    const int b    = tile / (T_ / 16);
    const int mt   = tile - b * (T_ / 16);

    const float* Arow = phi + ((size_t)(b * T_ + mt * 16 + lid)) * U_;
    const float* Ob   = onehot + (size_t)b * U_ * V_;

    v8f acc[5] = {};
    for (int k0 = 0; k0 < U_; k0 += 4) {
        // A fragment: lane holds row M=lid, K = {k0+2*half, k0+2*half+1}
        const int ka = k0 + 2 * half;
        const v2f a  = *(const v2f*)(Arow + ka);     // 8B-aligned (ka even)
        const float* bp = Ob + (size_t)ka * V_ + lid;
        #pragma unroll
        for (int nt = 0; nt < 5; ++nt) {
            // B fragment: lane holds col N = nt*16+lid, K rows {ka, ka+1}
            v2f bf;
            bf.x = bp[nt * 16];
            bf.y = bp[nt * 16 + V_];
            acc[nt] = __builtin_amdgcn_wmma_f32_16x16x4_f32(
                false, a, false, bf, (short)0, acc[nt], false, false);
        }
    }

    // C/D layout: VGPR r, row M = r + 8*half, col N = nt*16 + lid
    float* Wrow = win + ((size_t)(b * T_ + mt * 16)) * V_;
    #pragma unroll
    for (int nt = 0; nt < 5; ++nt) {
        #pragma unroll
        for (int r = 0; r < 8; ++r) {
            Wrow[(size_t)(r + 8 * half) * V_ + nt * 16 + lid] = acc[nt][r];
        }
    }
}

// ------------------------------------------------------------------ launcher
extern "C" void kernel_launch(void* const* d_in, const int* in_sizes, int n_in,
                              void* d_out, int out_size, void* d_ws, size_t ws_size,
                              hipStream_t stream)
{
    const float* x      = (const float*)d_in[0];
    const float* onehot = (const float*)d_in[1];
    const float* Wa     = (const float*)d_in[2];
    const float* ba     = (const float*)d_in[3];
    const float* Wb     = (const float*)d_in[4];
    const float* bb     = (const float*)d_in[5];
    const float* Wk     = (const float*)d_in[6];
    const float* bk     = (const float*)d_in[7];

    float* window = (float*)d_out;                       // (B,T,V)
    float* phi    = (float*)d_out + (size_t)B_ * T_ * V_; // (B,T,U)

    proj_kernel<<<B_ * T_, 32, 0, stream>>>(x, Wa, ba, Wb, bb, Wk, bk, phi);
    scan_kernel<<<B_ * K_, T_, 0, stream>>>(phi);
    phi_kernel<<<B_ * T_, 256, 0, stream>>>(phi);
    window_kernel<<<(B_ * T_ / 16) / 4, 128, 0, stream>>>(phi, onehot, window);

    (void)in_sizes; (void)n_in; (void)out_size; (void)d_ws; (void)ws_size;
}